// WordMajority_45655502357335
// MI455X (gfx1250) — compile-verified
//
#include <hip/hip_runtime.h>
#include <hip/hip_bf16.h>

// Memory-bound gather/argmax/one-hot. No matmul -> no WMMA; optimize the
// data path instead: b128 loads/stores, NT store hints for the 84MB of
// write-once output, and a global_prefetch warm-up of the 1.6MB table so
// gathers hit L2 (192MB) from the start.

typedef float v4f __attribute__((ext_vector_type(4)));
typedef int   v4i __attribute__((ext_vector_type(4)));

#define NTOK (256 * 8192)   // 2,097,152 tokens
#define NC   8              // classes
#define VOCAB 50257
#define TPT  4              // tokens per thread (-> 128B logits burst/thread)
#define BLK  256            // 8 wave32 per block

__global__ __launch_bounds__(BLK)
void WordMajority_kernel(const int*   __restrict__ x,
                         const int*   __restrict__ y,
                         const float* __restrict__ counts,
                         float*       __restrict__ out)
{
    const size_t tid  = (size_t)blockIdx.x * BLK + threadIdx.x;
    const size_t base = tid * TPT;            // first token for this thread

    // ---- warm L2 with the heavily-reused counts table ------------------
    // 50257*8*4B = 1.57MB = ~12565 cachelines of 128B; one prefetch per
    // line from the first ~12.6k threads. Emits global_prefetch_b8.
    {
        const size_t tableBytes = (size_t)VOCAB * NC * sizeof(float);
        const size_t nlines = (tableBytes + 127) / 128;
        if (tid < nlines) {
            __builtin_prefetch((const char*)counts + tid * 128, /*rw=*/0, /*loc=*/2);
        }
    }

    // ---- coalesced b128 loads of 4 token ids + 4 labels ----------------
    const v4i xv = *(const v4i*)(x + base);
    const v4i yv = *(const v4i*)(y + base);

    float* logits = out;                                // [NTOK*NC]
    float* y_out  = out + (size_t)NTOK * NC;            // [NTOK]
    float* yhat   = y_out + NTOK;                       // [NTOK]

    v4f yo, po;

    #pragma unroll
    for (int t = 0; t < TPT; ++t) {
        const int tok = xv[t];

        // gather one 32B row: two b128 loads (rows are 32B aligned)
        const v4f* row = (const v4f*)(counts + (size_t)tok * NC);
        const v4f lo = row[0];
        const v4f hi = row[1];

        const float v[NC] = { lo.x, lo.y, lo.z, lo.w,
                              hi.x, hi.y, hi.z, hi.w };

        // argmax, first-max-wins (matches jnp.argmax tie-breaking)
        int   best = 0;
        float bv   = v[0];
        #pragma unroll
        for (int i = 1; i < NC; ++i) {
            if (v[i] > bv) { bv = v[i]; best = i; }
        }

        // one-hot row, stored non-temporally as two b128s
        v4f o0, o1;
        #pragma unroll
        for (int i = 0; i < 4; ++i) {
            o0[i] = (best == i    ) ? 1.0f : 0.0f;
            o1[i] = (best == i + 4) ? 1.0f : 0.0f;
        }
        v4f* dst = (v4f*)(logits + (base + (size_t)t) * NC);
        __builtin_nontemporal_store(o0, dst);
        __builtin_nontemporal_store(o1, dst + 1);

        yo[t] = (float)yv[t];
        po[t] = (float)best;
    }

    // packed 16B NT stores for the pass-through label and prediction
    __builtin_nontemporal_store(yo, (v4f*)(y_out + base));
    __builtin_nontemporal_store(po, (v4f*)(yhat + base));
}

extern "C" void kernel_launch(void* const* d_in, const int* in_sizes, int n_in,
                              void* d_out, int out_size, void* d_ws, size_t ws_size,
                              hipStream_t stream)
{
    const int*   x      = (const int*)  d_in[0];   // [256,8192] int32 token ids
    const int*   y      = (const int*)  d_in[1];   // [256,8192] int32 labels
    const float* counts = (const float*)d_in[2];   // [50257,8] fp32
    float*       out    = (float*)      d_out;     // logits ‖ y ‖ y_hat

    const int nblocks = NTOK / (BLK * TPT);        // 2048 blocks, 8 waves each
    WordMajority_kernel<<<nblocks, BLK, 0, stream>>>(x, y, counts, out);
}